// SuperpixelGCN_57681410785898
// MI455X (gfx1250) — compile-verified
//
#include <hip/hip_runtime.h>
#include <hip/hip_bf16.h>

typedef float v2f __attribute__((ext_vector_type(2)));
typedef float v4f __attribute__((ext_vector_type(4)));
typedef float v8f __attribute__((ext_vector_type(8)));

// ---------------------------------------------------------------------------
// Zero-fill int32 (graph-capture-safe)
// ---------------------------------------------------------------------------
__global__ void zero_i32(int* __restrict__ p, size_t n) {
    size_t i = (size_t)blockIdx.x * blockDim.x + threadIdx.x;
    size_t stride = (size_t)gridDim.x * blockDim.x;
    for (; i < n; i += stride) p[i] = 0;
}

// ---------------------------------------------------------------------------
// CSR build: in-degree counts per destination node
// ---------------------------------------------------------------------------
__global__ void csr_count(const long long* __restrict__ col, int* __restrict__ cnt, int E) {
    int e = blockIdx.x * blockDim.x + threadIdx.x;
    if (e < E) atomicAdd(&cnt[(int)col[e]], 1);
}

// dinv = rsqrt(cnt + 1)   (self-loop included)
__global__ void dinv_from_cnt(const int* __restrict__ cnt, float* __restrict__ dinv, int N) {
    int i = blockIdx.x * blockDim.x + threadIdx.x;
    if (i < N) dinv[i] = rsqrtf((float)cnt[i] + 1.0f);
}

// ---------------------------------------------------------------------------
// 3-kernel exclusive scan over cnt[N]  (N <= 65536: one level of block sums)
// ---------------------------------------------------------------------------
__global__ void scan_block(const int* __restrict__ cnt, int* __restrict__ excl,
                           int* __restrict__ bsum, int N) {
    __shared__ int tmp[256];
    const int t = threadIdx.x;
    const int i = blockIdx.x * 256 + t;
    const int v = (i < N) ? cnt[i] : 0;
    tmp[t] = v;
    __syncthreads();
    for (int off = 1; off < 256; off <<= 1) {
        int u = (t >= off) ? tmp[t - off] : 0;
        __syncthreads();
        tmp[t] += u;
        __syncthreads();
    }
    if (i < N) excl[i] = tmp[t] - v;              // exclusive prefix within block
    if (t == 255) bsum[blockIdx.x] = tmp[255];    // block total
}

__global__ void scan_top(int* __restrict__ bsum, int B) {
    __shared__ int tmp[256];
    const int t = threadIdx.x;
    const int v = (t < B) ? bsum[t] : 0;
    tmp[t] = v;
    __syncthreads();
    for (int off = 1; off < 256; off <<= 1) {
        int u = (t >= off) ? tmp[t - off] : 0;
        __syncthreads();
        tmp[t] += u;
        __syncthreads();
    }
    if (t < B) bsum[t] = tmp[t] - v;              // exclusive
}

__global__ void scan_add(int* __restrict__ excl, const int* __restrict__ bsum,
                         int* __restrict__ cursor, int N) {
    const int i = blockIdx.x * 256 + threadIdx.x;
    if (i < N) {
        const int v = excl[i] + bsum[blockIdx.x];
        excl[i]   = v;
        cursor[i] = v;
    }
}

// Fill CSR adjacency: srcIdx[offset[col] ...] = row
__global__ void csr_fill(const long long* __restrict__ row, const long long* __restrict__ col,
                         int* __restrict__ cursor, int* __restrict__ srcIdx, int E) {
    int e = blockIdx.x * blockDim.x + threadIdx.x;
    if (e < E) {
        const int c = (int)col[e];
        const int p = atomicAdd(&cursor[c], 1);
        srcIdx[p] = (int)row[e];
    }
}

// ---------------------------------------------------------------------------
// WMMA fp32 GEMM: C[M,N] = A[M,K] @ B[K,N].
// blockDim = 256 (8 waves). Each wave computes a 16x32 tile (two 16x16 WMMAs).
// N == 256, M % 16 == 0, K % 4 == 0. EXEC all-ones everywhere (WMMA req).
// ---------------------------------------------------------------------------
__global__ void gemm_wmma_f32(const float* __restrict__ A, const float* __restrict__ B,
                              float* __restrict__ C, int M, int N, int K) {
    const int lane = threadIdx.x & 31;
    const int wave = threadIdx.x >> 5;
    const int m0 = blockIdx.x * 16;

    const int mrow  = m0 + (lane & 15);
    const int khalf = (lane >> 4) << 1;            // 0 or 2
    const int ncol0 = wave * 32 + (lane & 15);
    const int ncol1 = ncol0 + 16;

    v8f acc0 = {};
    v8f acc1 = {};

    for (int k0 = 0; k0 < K; k0 += 4) {
        const int kb = k0 + khalf;
        v2f a;
        a.x = A[(size_t)mrow * K + kb];
        a.y = A[(size_t)mrow * K + kb + 1];
        v2f b0;
        b0.x = B[(size_t)kb * N + ncol0];
        b0.y = B[(size_t)(kb + 1) * N + ncol0];
        v2f b1;
        b1.x = B[(size_t)kb * N + ncol1];
        b1.y = B[(size_t)(kb + 1) * N + ncol1];
        acc0 = __builtin_amdgcn_wmma_f32_16x16x4_f32(false, a, false, b0, (short)0, acc0,
                                                     false, false);
        acc1 = __builtin_amdgcn_wmma_f32_16x16x4_f32(false, a, false, b1, (short)0, acc1,
                                                     false, false);
    }

    const int mbase = m0 + ((lane >> 4) << 3);     // lanes 16-31 hold M = i+8
    const int nlo = lane & 15;
#pragma unroll
    for (int i = 0; i < 8; ++i) {
        C[(size_t)(mbase + i) * N + wave * 32 + nlo]      = acc0[i];
        C[(size_t)(mbase + i) * N + wave * 32 + 16 + nlo] = acc1[i];
    }
}

// ---------------------------------------------------------------------------
// Fused gather aggregation + self-loop + bias + ReLU (no atomics):
//   out[n][c] = relu( dinv[n] * sum_{s in adj(n)} dinv[s]*h[s][c]
//                     + h[n][c]*dinv[n]^2 + b[c] )
// One wave per node. Lane owns channels [lane*8, lane*8+8): two b128 loads
// per visited row per lane (contiguous 1KB per wave, fully coalesced).
// Edge loop unrolled 2x so both rows' b128 loads are in flight together.
// ---------------------------------------------------------------------------
__global__ void gather_combine_relu(const float* __restrict__ h, const int* __restrict__ off,
                                    const int* __restrict__ cnt, const int* __restrict__ srcIdx,
                                    const float* __restrict__ dinv, const float* __restrict__ b,
                                    float* __restrict__ out, int N) {
    const int n = blockIdx.x * (blockDim.x >> 5) + (threadIdx.x >> 5);
    if (n >= N) return;
    const int lane = threadIdx.x & 31;
    const int cbase = lane * 8;
    const int beg = off[n];
    const int num = cnt[n];

    v4f accA = {}, accB = {};

    int k = 0;
    for (; k + 1 < num; k += 2) {
        const int s0 = srcIdx[beg + k];
        const int s1 = srcIdx[beg + k + 1];
        const float w0 = dinv[s0];
        const float w1 = dinv[s1];
        const v4f* __restrict__ p0 = (const v4f*)(h + (size_t)s0 * 256 + cbase);
        const v4f* __restrict__ p1 = (const v4f*)(h + (size_t)s1 * 256 + cbase);
        const v4f a0 = p0[0];
        const v4f b0 = p0[1];
        const v4f a1 = p1[0];
        const v4f b1 = p1[1];
        accA += a0 * w0;
        accB += b0 * w0;
        accA += a1 * w1;
        accB += b1 * w1;
    }
    if (k < num) {
        const int s0 = srcIdx[beg + k];
        const float w0 = dinv[s0];
        const v4f* __restrict__ p0 = (const v4f*)(h + (size_t)s0 * 256 + cbase);
        accA += p0[0] * w0;
        accB += p0[1] * w0;
    }

    const float dn  = dinv[n];
    const float dn2 = dn * dn;
    const v4f* __restrict__ hn = (const v4f*)(h + (size_t)n * 256 + cbase);
    const v4f* __restrict__ bb = (const v4f*)(b + cbase);
    const v4f hnA = hn[0], hnB = hn[1];
    const v4f bA = bb[0], bB = bb[1];

    v4f oA = accA * dn + hnA * dn2 + bA;
    v4f oB = accB * dn + hnB * dn2 + bB;
#pragma unroll
    for (int j = 0; j < 4; ++j) {
        oA[j] = oA[j] > 0.0f ? oA[j] : 0.0f;
        oB[j] = oB[j] > 0.0f ? oB[j] : 0.0f;
    }
    v4f* __restrict__ po = (v4f*)(out + (size_t)n * 256 + cbase);
    po[0] = oA;
    po[1] = oB;
}

// ---------------------------------------------------------------------------
// Global mean pool, atomic-free: batch is sorted, so graph g's nodes form a
// contiguous range found by binary search. One block (256 threads = channels)
// per graph.
// ---------------------------------------------------------------------------
__device__ __forceinline__ int lower_bound_ll(const long long* __restrict__ a, int n,
                                              long long key) {
    int lo = 0, hi = n;
    while (lo < hi) {
        int mid = (lo + hi) >> 1;
        if (a[mid] < key) lo = mid + 1; else hi = mid;
    }
    return lo;
}

__global__ void pool_mean(const float* __restrict__ h, const long long* __restrict__ batch,
                          float* __restrict__ gmean, int N, int C) {
    const int g = blockIdx.x;
    const int c = threadIdx.x;          // blockDim == C == 256
    const int lo = lower_bound_ll(batch, N, (long long)g);
    const int hi = lower_bound_ll(batch, N, (long long)g + 1);
    float acc = 0.0f;
    for (int n = lo; n < hi; ++n)
        acc += h[(size_t)n * C + c];
    float cntf = (float)(hi - lo);
    cntf = cntf > 1.0f ? cntf : 1.0f;
    gmean[(size_t)g * C + c] = acc / cntf;
}

// ---------------------------------------------------------------------------
// g2 = relu(gmean @ Wl1 + bl1)   [G x 64]; one block per graph
// ---------------------------------------------------------------------------
__global__ void mlp1(const float* __restrict__ gmean, const float* __restrict__ Wl1,
                     const float* __restrict__ bl1, float* __restrict__ g2, int C, int H) {
    const int g = blockIdx.x;
    const int t = threadIdx.x;          // blockDim == H == 64
    float acc = bl1[t];
    const float* __restrict__ gr = gmean + (size_t)g * C;
    for (int k = 0; k < C; ++k)
        acc += gr[k] * Wl1[(size_t)k * H + t];
    g2[(size_t)g * H + t] = acc > 0.0f ? acc : 0.0f;
}

// ---------------------------------------------------------------------------
// out = log_softmax(g2 @ Wl2 + bl2)   [G x 10]; one thread per graph
// ---------------------------------------------------------------------------
__global__ void mlp2_lsm(const float* __restrict__ g2, const float* __restrict__ Wl2,
                         const float* __restrict__ bl2, float* __restrict__ out,
                         int G, int H, int NCLS) {
    const int g = blockIdx.x * blockDim.x + threadIdx.x;
    if (g >= G) return;
    float z[10];
    const float* __restrict__ gr = g2 + (size_t)g * H;
    for (int j = 0; j < NCLS; ++j) {
        float acc = bl2[j];
        for (int k = 0; k < H; ++k)
            acc += gr[k] * Wl2[(size_t)k * NCLS + j];
        z[j] = acc;
    }
    float mx = z[0];
    for (int j = 1; j < NCLS; ++j) mx = z[j] > mx ? z[j] : mx;
    float s = 0.0f;
    for (int j = 0; j < NCLS; ++j) s += expf(z[j] - mx);
    const float lse = mx + logf(s);
    for (int j = 0; j < NCLS; ++j)
        out[(size_t)g * NCLS + j] = z[j] - lse;
}

// ---------------------------------------------------------------------------
extern "C" void kernel_launch(void* const* d_in, const int* in_sizes, int n_in,
                              void* d_out, int out_size, void* d_ws, size_t ws_size,
                              hipStream_t stream) {
    const float*     x     = (const float*)d_in[0];
    const long long* ei    = (const long long*)d_in[1];   // int64 [2, E]
    const long long* batch = (const long long*)d_in[2];   // int64 [N] (sorted)
    const float*     W1    = (const float*)d_in[3];
    const float*     b1    = (const float*)d_in[4];
    const float*     W2    = (const float*)d_in[5];
    const float*     b2    = (const float*)d_in[6];
    const float*     Wl1   = (const float*)d_in[7];
    const float*     bl1   = (const float*)d_in[8];
    const float*     Wl2   = (const float*)d_in[9];
    const float*     bl2   = (const float*)d_in[10];

    const int IN_C = 128, HID = 256, G = 128, H2 = 64, NCLS = 10;
    const int N = in_sizes[0] / IN_C;     // 50000 (multiple of 16, <= 65536)
    const int E = in_sizes[1] / 2;        // 1600000

    const long long* row = ei;
    const long long* col = ei + E;

    const int NB = (N + 255) / 256;       // scan blocks (<= 256)

    // ---- workspace layout --------------------------------------------------
    char* base = (char*)d_ws;
    int* cnt    = (int*)base;                        base += (size_t)N * sizeof(int);
    int* off    = (int*)base;                        base += (size_t)N * sizeof(int);
    int* cursor = (int*)base;                        base += (size_t)N * sizeof(int);
    int* bsum   = (int*)base;                        base += 256 * sizeof(int);
    int* srcIdx = (int*)base;                        base += (size_t)E * sizeof(int);
    base = (char*)(((uintptr_t)base + 255) & ~(uintptr_t)255);
    float* dinv  = (float*)base;                     base += (size_t)N * sizeof(float);
    base = (char*)(((uintptr_t)base + 255) & ~(uintptr_t)255);
    float* buf0  = (float*)base;                     base += (size_t)N * HID * sizeof(float);
    float* buf1  = (float*)base;                     base += (size_t)N * HID * sizeof(float);
    float* gmean = (float*)base;                     base += (size_t)G * HID * sizeof(float);
    float* g2    = (float*)base;

    // ---- CSR build (atomic-light) ------------------------------------------
    zero_i32<<<256, 256, 0, stream>>>(cnt, (size_t)N);
    csr_count<<<(E + 255) / 256, 256, 0, stream>>>(col, cnt, E);
    dinv_from_cnt<<<(N + 255) / 256, 256, 0, stream>>>(cnt, dinv, N);
    scan_block<<<NB, 256, 0, stream>>>(cnt, off, bsum, N);
    scan_top<<<1, 256, 0, stream>>>(bsum, NB);
    scan_add<<<NB, 256, 0, stream>>>(off, bsum, cursor, N);
    csr_fill<<<(E + 255) / 256, 256, 0, stream>>>(row, col, cursor, srcIdx, E);

    // ---- layer 1: buf0 = x @ W1 ; gather+combine -> buf1 -------------------
    gemm_wmma_f32<<<N / 16, 256, 0, stream>>>(x, W1, buf0, N, HID, IN_C);
    gather_combine_relu<<<(N + 7) / 8, 256, 0, stream>>>(buf0, off, cnt, srcIdx, dinv, b1,
                                                         buf1, N);

    // ---- layer 2: buf0 = buf1 @ W2 ; gather+combine -> buf1 ----------------
    gemm_wmma_f32<<<N / 16, 256, 0, stream>>>(buf1, W2, buf0, N, HID, HID);
    gather_combine_relu<<<(N + 7) / 8, 256, 0, stream>>>(buf0, off, cnt, srcIdx, dinv, b2,
                                                         buf1, N);

    // ---- pool + MLP head ---------------------------------------------------
    pool_mean<<<G, HID, 0, stream>>>(buf1, batch, gmean, N, HID);
    mlp1<<<G, H2, 0, stream>>>(gmean, Wl1, bl1, g2, HID, H2);
    mlp2_lsm<<<1, G, 0, stream>>>(g2, Wl2, bl2, (float*)d_out, G, H2, NCLS);
}